// CP_Based_67602785239312
// MI455X (gfx1250) — compile-verified
//
#include <hip/hip_runtime.h>

// CP-decomposition layer: out[b,u] = sum_r prod_f ( x0[b,f]*K[0,r,f,u] + x1[b,f]*K[1,r,f,u] )
// with x0 = 1/sqrt(1+X^2), x1 = X/sqrt(1+X^2).
//
// WMMA mapping: for each feature f, T[16b x 16c] = A[16b x 4] * Bf[4 x 16c]
// (K=2 real, padded to 4 with zeros), c = r*8+u (80 cols = 5 tiles of 16).
// D-tile lane layout is f-invariant, so prod over f = elementwise product of
// successive WMMA results held in 8 accumulator VGPRs.

typedef float v2f __attribute__((ext_vector_type(2)));
typedef float v8f __attribute__((ext_vector_type(8)));

#define F_DIM   32
#define RANK    10
#define UNITS   8
#define NTILE   5          // 80 columns / 16
#define ROWS_PER_BLOCK 64  // 4 waves * 16 rows
#define THREADS 128        // 4 wave32

__global__ __launch_bounds__(THREADS)
void cp_wmma_kernel(const float* __restrict__ X,
                    const float* __restrict__ K,
                    float* __restrict__ out) {
  // {x0,x1} per (row,f); row padded to 33 float2 to avoid 256B bank period
  __shared__ float2 sX[ROWS_PER_BLOCK][F_DIM + 1];
  // {k0,k1} per (f, tile, col<16); K-rows 2..3 of the WMMA B are zero via sZero
  __shared__ float2 sB[F_DIM][NTILE][16];
  // per-wave D staging for the rank reduction
  __shared__ float  sP[THREADS / 32][16][16];
  __shared__ float2 sZero;

  const int tid = threadIdx.x;
  const int rowBaseG = blockIdx.x * ROWS_PER_BLOCK;

  // ---- stage kernel fragments ----
  if (tid == 0) sZero = make_float2(0.0f, 0.0f);
  for (int i = tid; i < F_DIM * NTILE * 16; i += THREADS) {
    const int f = i / (NTILE * 16);
    const int rem = i % (NTILE * 16);
    const int t = rem >> 4, n = rem & 15;
    const int c = t * 16 + n;
    const int r = c / UNITS, u = c % UNITS;
    // kernel layout [2, RANK, F, UNITS] row-major
    const float k0 = K[((0 * RANK + r) * F_DIM + f) * UNITS + u];
    const float k1 = K[((1 * RANK + r) * F_DIM + f) * UNITS + u];
    sB[f][t][n] = make_float2(k0, k1);
  }
  // ---- stage normalized X powers ----
  for (int i = tid; i < ROWS_PER_BLOCK * F_DIM; i += THREADS) {
    const int row = i / F_DIM, f = i % F_DIM;
    const float xv = X[(rowBaseG + row) * F_DIM + f];
    const float inv = 1.0f / sqrtf(fmaf(xv, xv, 1.0f));
    sX[row][f] = make_float2(inv, xv * inv);
  }
  __syncthreads();

  const int wave = tid >> 5;
  const int lane = tid & 31;
  const int rowBase = wave * 16;          // 16 rows per wave
  const bool lo = (lane < 16);

  // A-fragment source: lanes 0..15 -> real {x0,x1}; lanes 16..31 -> zeros (K=2..3 pad)
  const float2* aBase = lo ? &sX[rowBase + lane][0] : &sZero;
  const int aStride = lo ? 1 : 0;

  float acc[4] = {0.0f, 0.0f, 0.0f, 0.0f};

  for (int t = 0; t < NTILE; ++t) {
    const float2* bBase = lo ? &sB[0][t][lane] : &sZero;
    const int bStride = lo ? (NTILE * 16) : 0;   // float2 stride per f

    v8f P;
    #pragma unroll
    for (int j = 0; j < 8; ++j) P[j] = 1.0f;
    v8f zero8 = {};

    #pragma unroll 8
    for (int f = 0; f < F_DIM; ++f) {
      const float2 af = aBase[f * aStride];
      const float2 bf = bBase[f * bStride];
      v2f A;  A[0] = af.x;  A[1] = af.y;
      v2f Bv; Bv[0] = bf.x; Bv[1] = bf.y;
      // D = A(16x4) * B(4x16) + 0
      v8f D = __builtin_amdgcn_wmma_f32_16x16x4_f32(
          false, A, false, Bv, (short)0, zero8, false, false);
      P *= D;   // elementwise product over features (pk_mul-able)
    }

    // stage D-layout tile: VGPR j -> (M = j or j+8, N = lane&15)
    const int m0 = lo ? 0 : 8;
    const int n  = lane & 15;
    #pragma unroll
    for (int j = 0; j < 8; ++j) sP[wave][m0 + j][n] = P[j];
    // intra-wave LDS ops are in-order; no barrier needed

    // tile t holds columns c = 16t..16t+15 -> ranks {2t, 2t+1}, units 0..7
    #pragma unroll
    for (int i = 0; i < 4; ++i) {
      const int p = lane + 32 * i;       // 0..127 -> (row, unit)
      const int mm = p >> 3, u = p & 7;
      acc[i] += sP[wave][mm][u] + sP[wave][mm][8 + u];
    }
  }

  // ---- store out[b, u] ----
  #pragma unroll
  for (int i = 0; i < 4; ++i) {
    const int p = lane + 32 * i;
    const int mm = p >> 3, u = p & 7;
    const int b = rowBaseG + rowBase + mm;
    out[b * UNITS + u] = acc[i];
  }
}

extern "C" void kernel_launch(void* const* d_in, const int* in_sizes, int n_in,
                              void* d_out, int out_size, void* d_ws, size_t ws_size,
                              hipStream_t stream) {
  const float* X = (const float*)d_in[0];   // [131072, 32] f32
  const float* K = (const float*)d_in[1];   // [2, 10, 32, 8] f32
  float* out = (float*)d_out;               // [131072, 8] f32

  const int B = in_sizes[0] / F_DIM;        // 131072
  const int grid = B / ROWS_PER_BLOCK;      // 2048
  cp_wmma_kernel<<<grid, THREADS, 0, stream>>>(X, K, out);
}